// Projection_34926674051155
// MI455X (gfx1250) — compile-verified
//
#include <hip/hip_runtime.h>

typedef __attribute__((ext_vector_type(2))) float v2f;
typedef __attribute__((ext_vector_type(4))) float v4f;
typedef __attribute__((ext_vector_type(8))) float v8f;

#define N_IMG  512
#define N_Z    8
#define N_VIEW 360
#define N_DET  512
#define N_S    512
#define NN     (N_IMG * N_IMG)

// One wave = one (view, 16-detector tile), all 8 z slices.
// Lanes 0-15 / 16-31 are detectors 0..15 feeding K-pairs {0,1} / {2,3} of the
// 16x4 f32 A-matrix of V_WMMA_F32_16X16X4_F32; B = ones -> every column of D
// accumulates sum_s(sample) per detector (the line integral), one
// accumulator per z.
__global__ __launch_bounds__(256) void fp_wmma_kernel(const float* __restrict__ vol,
                                                      float* __restrict__ out) {
  const int wave  = (blockIdx.x << 3) + (threadIdx.x >> 5);
  const int lane  = threadIdx.x & 31;
  const int view  = wave >> 5;   // 32 detector tiles per view
  const int dtile = wave & 31;

  // f32 angle exactly like the reference, then high-precision trig of it.
  const float theta = (float)view * 0.00872664625997164788f;
  const float c  = (float)cos((double)theta);
  const float si = (float)sin((double)theta);

  const int   d     = (dtile << 4) + (lane & 15);
  const float t     = (float)d - 255.5f;          // (N_DET-1)/2
  const int   khalf = (lane >> 4) << 1;           // 0 for lanes 0-15, 2 for 16-31

  const float bx = fmaf(-si, t, 255.5f);          // ix = c*sv + bx
  const float by = fmaf( c,  t, 255.5f);          // iy = si*sv + by

  const v8f vzero = {0.f, 0.f, 0.f, 0.f, 0.f, 0.f, 0.f, 0.f};
  v8f acc[N_Z];
#pragma unroll
  for (int z = 0; z < N_Z; ++z) acc[z] = vzero;

  const v2f bones = {1.0f, 1.0f};                 // B = ones(4x16), layout-free

  for (int it = 0; it < N_S / 4; ++it) {
    const int sbase = (it << 2) + khalf;          // this lane's two K slots

    float w00[2], w01[2], w10[2], w11[2];
    int   a0[2], a1[2];
#pragma unroll
    for (int u = 0; u < 2; ++u) {
      const float sv = (float)(sbase + u) - 255.5f;   // s coordinate (DS=1)
      const float ix = fmaf(c,  sv, bx);
      const float iy = fmaf(si, sv, by);
      // Tent weights around a clamped 2x2 footprint: for in-bounds samples
      // these are exactly (1-fx, fx)x(1-fy, fy); any pixel outside [0,511]
      // gets weight 0, which reproduces mode='constant', cval=0 exactly.
      int xb = (int)floorf(ix); xb = min(max(xb, 0), N_IMG - 2);
      int yb = (int)floorf(iy); yb = min(max(yb, 0), N_IMG - 2);
      const float wxa = fmaxf(0.0f, 1.0f - fabsf(ix - (float)xb));
      const float wxb = fmaxf(0.0f, 1.0f - fabsf(ix - (float)(xb + 1)));
      const float wya = fmaxf(0.0f, 1.0f - fabsf(iy - (float)yb));
      const float wyb = fmaxf(0.0f, 1.0f - fabsf(iy - (float)(yb + 1)));
      w00[u] = wxa * wya;  w01[u] = wxb * wya;
      w10[u] = wxa * wyb;  w11[u] = wxb * wyb;
      a0[u] = yb * N_IMG + xb;        // row yb, cols xb..xb+1  (always in-bounds)
      a1[u] = a0[u] + N_IMG;          // row yb+1, cols xb..xb+1
    }

    // Gather phase: all 8z x 2 samples x 4 corners into distinct registers so
    // the scheduler keeps ~32 loads in flight (adjacent pairs merge to b64).
    float g[N_Z][2][4];
#pragma unroll
    for (int z = 0; z < N_Z; ++z) {
      const float* __restrict__ p = vol + z * NN;
#pragma unroll
      for (int u = 0; u < 2; ++u) {
        g[z][u][0] = p[a0[u]];
        g[z][u][1] = p[a0[u] + 1];
        g[z][u][2] = p[a1[u]];
        g[z][u][3] = p[a1[u] + 1];
      }
    }

    // Combine + K-reduce via WMMA, one accumulator per z.
#pragma unroll
    for (int z = 0; z < N_Z; ++z) {
      const float s0 = fmaf(w11[0], g[z][0][3],
                       fmaf(w10[0], g[z][0][2],
                       fmaf(w01[0], g[z][0][1], w00[0] * g[z][0][0])));
      const float s1 = fmaf(w11[1], g[z][1][3],
                       fmaf(w10[1], g[z][1][2],
                       fmaf(w01[1], g[z][1][1], w00[1] * g[z][1][0])));
      v2f av = {s0, s1};  // A VGPR0 -> K=khalf, VGPR1 -> K=khalf+1
      acc[z] = __builtin_amdgcn_wmma_f32_16x16x4_f32(
          /*neg_a=*/false, av, /*neg_b=*/false, bones,
          /*c_mod=*/(short)0, acc[z], /*reuse_a=*/false, /*reuse_b=*/false);
    }
  }

  // B is all-ones, so every column of D is identical: acc[z][r] holds the
  // detector-r sum in ALL lanes 0-15 and the detector-(8+r) sum in ALL lanes
  // 16-31. One branch: lanes 0 and 16 store 8 floats each as two b128s.
  if ((lane & 15) == 0) {
    const int hi = (lane >> 4) << 3;              // 0 for lane 0, 8 for lane 16
    float* o = out + (size_t)view * N_DET + (dtile << 4) + hi;
#pragma unroll
    for (int z = 0; z < N_Z; ++z) {
      v4f lo4 = {acc[z][0], acc[z][1], acc[z][2], acc[z][3]};
      v4f hi4 = {acc[z][4], acc[z][5], acc[z][6], acc[z][7]};
      *(v4f*)(o + (size_t)z * (N_VIEW * N_DET))     = lo4;
      *(v4f*)(o + (size_t)z * (N_VIEW * N_DET) + 4) = hi4;
    }
  }
}

extern "C" void kernel_launch(void* const* d_in, const int* in_sizes, int n_in,
                              void* d_out, int out_size, void* d_ws, size_t ws_size,
                              hipStream_t stream) {
  (void)in_sizes; (void)n_in; (void)out_size; (void)d_ws; (void)ws_size;
  const float* vol = (const float*)d_in[0];   // (8,1,512,512) f32, flat
  float* out = (float*)d_out;                 // (8,1,360,512) f32, flat
  // 360 views * 32 det-tiles = 11520 waves = 1440 blocks of 8 waves
  fp_wmma_kernel<<<dim3(1440), dim3(256), 0, stream>>>(vol, out);
}